// ModelEncodeDecodeSimple_16166256902401
// MI455X (gfx1250) — compile-verified
//
#include <hip/hip_runtime.h>

// ---------------- constants / workspace layout (floats) ----------------
#define OFF_WS2 0            // W_S2[60][100]                       : 6000
#define OFF_F   6000         // Fr[20][20], Fi[20][20]              : 800
#define OFF_YH  6800         // yhat[(s*2+i)*5+o] re/im             : 2000
#define OFF_D   8800         // D[l][k][(m+l)*L+(n+l)]              : 26600
#define OFF_FH  35400        // fhat[b][ (s*2+f)*2 + c ]            : 512*400

typedef __attribute__((ext_vector_type(2))) float v2f;
typedef __attribute__((ext_vector_type(8))) float v8f;

__device__ __forceinline__ v8f wmma4(v2f a, v2f b, v8f c) {
  // D = A(16x4) * B(4x16) + C(16x16), fp32 WMMA on gfx1250
  return __builtin_amdgcn_wmma_f32_16x16x4_f32(false, a, false, b, (short)0, c,
                                               false, false);
}

__device__ __forceinline__ int zcumf(int l) {  // sum_{j<l}(2j+1)^2
  return l * (2 * l - 1) * (2 * l + 1) / 3;
}

// ---------------- Wigner-d (double precision, exact for l<=9) ----------------
__device__ double dfact(int n) { double r = 1.0; for (int i = 2; i <= n; ++i) r *= (double)i; return r; }
__device__ double ipow(double x, int n) { double r = 1.0; for (int i = 0; i < n; ++i) r *= x; return r; }

__device__ double wigd(int l, int mp, int m, double beta) {
  double cb = cos(beta * 0.5), sb = sin(beta * 0.5);
  double pref = sqrt(dfact(l + m) * dfact(l - m) * dfact(l + mp) * dfact(l - mp));
  int s0 = (m - mp) > 0 ? (m - mp) : 0;
  int s1 = (l + m) < (l - mp) ? (l + m) : (l - mp);
  double tot = 0.0;
  for (int s = s0; s <= s1; ++s) {
    double den = dfact(l + m - s) * dfact(s) * dfact(mp - m + s) * dfact(l - mp - s);
    double sgn = ((mp - m + s) & 1) ? -1.0 : 1.0;
    tot += sgn / den * ipow(cb, 2 * l + m - mp - 2 * s) * ipow(sb, mp - m + 2 * s);
  }
  return pref * tot;
}

// ---------------- Kernel 1: build constant tables + yhat ----------------
__global__ void __launch_bounds__(256) precompute_kernel(const float* __restrict__ kern,
                                                         float* __restrict__ ws) {
  __shared__ double wq[60];
  const double PI = 3.14159265358979323846;
  int tid = threadIdx.x;
  if (tid < 60) {  // Driscoll-Healy quadrature weights (B_IN = 30)
    double beta = PI * (2 * tid + 1) / 120.0;
    double ssum = 0.0;
    for (int k = 0; k < 30; ++k)
      ssum += sin((double)(2 * tid + 1) * (2 * k + 1) * PI / 120.0) / (double)(2 * k + 1);
    wq[tid] = (2.0 / 30.0) * sin(beta) * ssum;
  }
  for (int t = tid; t < 400; t += 256) {  // 20-pt inverse-DFT twiddles e^{+2pi i u p/20}
    int u = t / 20, p = t % 20;
    double ang = 2.0 * PI * (double)(u * p) / 20.0;
    ws[OFF_F + t] = (float)cos(ang);
    ws[OFF_F + 400 + t] = (float)sin(ang);
  }
  __syncthreads();
  for (int t = tid; t < 6000; t += 256) {  // W_S2[k][s] = w[k]*d^l_{mp,0}(beta_k)
    int k = t / 100, s = t % 100;
    int l = 0; while ((l + 1) * (l + 1) <= s) ++l;
    int mp = s - l * l - l;
    double beta = PI * (2 * k + 1) / 120.0;
    ws[OFF_WS2 + t] = (float)(wq[k] * wigd(l, mp, 0, beta));
  }
  for (int t = tid; t < 26600; t += 256) {  // D[l][k][mp][m] = (2l+1)*d^l(beta'_k)
    int l = 0; while (20 * zcumf(l + 1) <= t) ++l;
    int r = t - 20 * zcumf(l);
    int L = 2 * l + 1;
    int k = r / (L * L);
    int rr = r % (L * L);
    int mp = rr / L - l, m = rr % L - l;
    double beta = PI * (2 * k + 1) / 40.0;
    ws[OFF_D + t] = (float)((double)(2 * l + 1) * wigd(l, mp, m, beta));
  }
  // yhat[s,i,o] = SCALING * sum_g kernel[i,o,g] * d^l_{mp,0}(pi/160) e^{-i mp GA_g}
  double SC = 1.0 / sqrt(6.0 * 2.0 * 10000.0 / 900.0);
  for (int t = tid; t < 1000; t += 256) {
    int s = t / 10, r = t % 10, i = r / 5, o = r % 5;
    int l = 0; while ((l + 1) * (l + 1) <= s) ++l;
    int mp = s - l * l - l;
    double d0 = wigd(l, mp, 0, PI / 160.0);
    double ar = 0.0, ai = 0.0;
    for (int g = 0; g < 6; ++g) {
      double ga = 2.0 * PI * (double)g / 6.0;
      double kv = (double)kern[(i * 5 + o) * 6 + g] * SC;
      ar += kv * d0 * cos((double)mp * ga);
      ai -= kv * d0 * sin((double)mp * ga);
    }
    ws[OFF_YH + t * 2 + 0] = (float)ar;
    ws[OFF_YH + t * 2 + 1] = (float)ai;
  }
}

// ---------------- Kernel 2: fhat[s,b,f] (DFT on 19 needed freqs + W_S2) ----------------
__global__ void __launch_bounds__(256) fhat_kernel(const float* __restrict__ x,
                                                   float* __restrict__ ws) {
  __shared__ float xs[7200];
  __shared__ float Ec[1140], Es[1140];
  __shared__ float Xr[2280], Xi[2280];
  const float PI_F = 3.14159265358979323846f;
  int b = blockIdx.x, tid = threadIdx.x;
  // b128 staging of x[b] (7200 floats = 1800 float4)
  const float4* xin4 = (const float4*)(x + (size_t)b * 7200u);
  float4* xs4 = (float4*)xs;
  for (int t = tid; t < 1800; t += 256) xs4[t] = xin4[t];
  for (int t = tid; t < 1140; t += 256) {  // E[j][mi] = e^{-2pi i j m/60}, m = mi-9
    int j = t / 19, mi = t % 19;
    float ang = -2.0f * PI_F * (float)(j * (mi - 9)) / 60.0f;
    Ec[t] = cosf(ang);
    Es[t] = sinf(ang);
  }
  __syncthreads();
  for (int t = tid; t < 2280; t += 256) {  // Xf19[f][k][mi] = sum_j x * E
    int f = t / 1140, r = t % 1140, k = r / 19, mi = r % 19;
    const float* row = &xs[f * 3600 + k * 60];
    float ar = 0.f, ai = 0.f;
#pragma unroll 4
    for (int j = 0; j < 60; ++j) {
      float xv = row[j];
      ar += xv * Ec[j * 19 + mi];
      ai += xv * Es[j * 19 + mi];
    }
    Xr[t] = ar;
    Xi[t] = ai;
  }
  __syncthreads();
  for (int t = tid; t < 200; t += 256) {  // fhat[s,f] = sum_k W_S2[k,s]*Xf19
    int s = t >> 1, f = t & 1;
    int l = 0; while ((l + 1) * (l + 1) <= s) ++l;
    int mi = (s - l * l - l) + 9;
    float ar = 0.f, ai = 0.f;
    for (int k = 0; k < 60; ++k) {
      float w = ws[OFF_WS2 + k * 100 + s];
      int xi = (f * 60 + k) * 19 + mi;
      ar += w * Xr[xi];
      ai += w * Xi[xi];
    }
    ws[OFF_FH + b * 400 + (s * 2 + f) * 2 + 0] = ar;
    ws[OFF_FH + b * 400 + (s * 2 + f) * 2 + 1] = ai;
  }
}

// ---------------- Kernel 3: per-(b,o) fused z -> T -> iDFT2 (WMMA f32) ----------------
__global__ void __launch_bounds__(256) s2conv_main_kernel(const float* __restrict__ ws,
                                                          const float* __restrict__ bias,
                                                          float* __restrict__ out) {
  __shared__ float fhs[400], yhs[400];
  // Transposed, zero-padded twiddles: Ft[p][u] = F[u][p], 32 rows (rows>=20 are 0)
  __shared__ float Ftr[640], Fti[640], Ftn[640];
  __shared__ float zRs[1330], zIs[1330];
  __shared__ float Trs[8000], Tis[8000];  // T[(k*20+u)*20+v]
  __shared__ float STr[8000], STi[8000];  // S^T[(k*20+q)*20+u]
  int tid = threadIdx.x;
  int lane = tid & 31, wave = tid >> 5;
  int bo = blockIdx.x;
  int b = bo / 5, o = bo % 5;
  float bval = bias[o];

  // Phase A: stage inputs, zero T
  for (int t = tid; t < 400; t += 256) fhs[t] = ws[OFF_FH + b * 400 + t];
  for (int t = tid; t < 400; t += 256) {
    int s = t >> 2, r = t & 3, i = r >> 1, c = r & 1;
    yhs[t] = ws[OFF_YH + ((s * 2 + i) * 5 + o) * 2 + c];
  }
  for (int t = tid; t < 640; t += 256) {  // padded transpose of F
    int p = t / 20, u = t % 20;
    float fr = (p < 20) ? ws[OFF_F + u * 20 + p] : 0.f;
    float fi = (p < 20) ? ws[OFF_F + 400 + u * 20 + p] : 0.f;
    Ftr[t] = fr;
    Fti[t] = fi;
    Ftn[t] = -fi;
  }
  for (int t = tid; t < 8000; t += 256) { Trs[t] = 0.f; Tis[t] = 0.f; }
  __syncthreads();

  // Phase B: z_l[m,n] = sum_i fhat[s_m,i] * conj(yhat[s_n,i,o])
  for (int t = tid; t < 1330; t += 256) {
    int l = 0; while (zcumf(l + 1) <= t) ++l;
    int L = 2 * l + 1;
    int r = t - zcumf(l);
    int sm = l * l + r / L, sn = l * l + r % L;
    float re = 0.f, im = 0.f;
#pragma unroll
    for (int i = 0; i < 2; ++i) {
      float fr = fhs[(sm * 2 + i) * 2], fi = fhs[(sm * 2 + i) * 2 + 1];
      float yr = yhs[sn * 4 + i * 2], yi = yhs[sn * 4 + i * 2 + 1];
      re += fr * yr + fi * yi;
      im += fi * yr - fr * yi;
    }
    zRs[t] = re;
    zIs[t] = im;
  }
  __syncthreads();

  // Phase C: T[k,u,v] = sum_l z_l[m,n] * D[l,k,m,n]   (u = m mod 20, v = n mod 20)
  for (int t = tid; t < 7220; t += 256) {
    int k = t / 361, r = t % 361;
    int m = r / 19 - 9, n = r % 19 - 9;
    int u = (m + 20) % 20, v = (n + 20) % 20;
    int am = m < 0 ? -m : m, an = n < 0 ? -n : n;
    int lmin = am > an ? am : an;
    float ar = 0.f, ai = 0.f;
    for (int l = lmin; l < 10; ++l) {
      int L = 2 * l + 1;
      int off = (m + l) * L + (n + l);
      float dv = ws[OFF_D + 20 * zcumf(l) + k * L * L + off];
      int zi = zcumf(l) + off;
      ar += zRs[zi] * dv;
      ai += zIs[zi] * dv;
    }
    Trs[(k * 20 + u) * 20 + v] = ar;
    Tis[(k * 20 + u) * 20 + v] = ai;
  }
  __syncthreads();

  // Phase D (WMMA): S[(k,u),q] = sum_v T[(k,u),v] * F[v,q]   M=400 N=20 K=20
  // Stored transposed: ST[(k,q),u]. All fragment loads are branch-free ds_load_b64.
  for (int tile = wave; tile < 50; tile += 8) {
    int mt = tile >> 1, nt = tile & 1;
    int mrow = mt * 16 + (lane & 15);
    int kb0 = (lane >> 4) * 2;
    int q = nt * 16 + (lane & 15);  // 0..31, rows >=20 of Ft are zero-padded
    v8f aSr = {}; v8f aSi = {};
#pragma unroll
    for (int ks = 0; ks < 5; ++ks) {
      int v0 = ks * 4 + kb0;                              // even
      v2f trA = *(const v2f*)&Trs[mrow * 20 + v0];
      v2f tiA = *(const v2f*)&Tis[mrow * 20 + v0];
      v2f frB = *(const v2f*)&Ftr[q * 20 + v0];           // F[v0][q], F[v0+1][q]
      v2f fiB = *(const v2f*)&Fti[q * 20 + v0];
      v2f fnB = *(const v2f*)&Ftn[q * 20 + v0];
      aSr = wmma4(trA, frB, aSr);   // + Tr*Fr
      aSr = wmma4(tiA, fnB, aSr);   // - Ti*Fi
      aSi = wmma4(trA, fiB, aSi);   // + Tr*Fi
      aSi = wmma4(tiA, frB, aSi);   // + Ti*Fr
    }
    if (q < 20) {
      // Transposed store: addr = k*400 + q*20 + u with (k,u) from Mr = Mr0 + j.
      // Single div; per-j address is base + j (+380 once when u wraps past 20).
      int Mr0 = mt * 16 + 8 * (lane >> 4);
      int k0 = Mr0 / 20, u0 = Mr0 - k0 * 20;
      int base = k0 * 400 + q * 20 + u0;
#pragma unroll
      for (int j = 0; j < 8; ++j) {
        int addr = base + j + ((u0 + j >= 20) ? 380 : 0);
        STr[addr] = aSr[j];
        STi[addr] = aSi[j];
      }
    }
  }
  __syncthreads();

  // Phase E (WMMA): out[p,(k,q)] = Re( sum_u F[u,p]*S[(k,u),q] ) + bias
  for (int tile = wave; tile < 50; tile += 8) {
    int pt = tile / 25, ct = tile % 25;
    int p = pt * 16 + (lane & 15);  // 0..31, Ft rows >=20 zero-padded
    int col = ct * 16 + (lane & 15);
    int kk = col / 20, qq = col - kk * 20;
    int kb0 = (lane >> 4) * 2;
    v8f acc = {};
#pragma unroll
    for (int ks = 0; ks < 5; ++ks) {
      int u0 = ks * 4 + kb0;                               // even
      v2f fA  = *(const v2f*)&Ftr[p * 20 + u0];            // F[u0][p], F[u0+1][p]
      v2f fnA = *(const v2f*)&Ftn[p * 20 + u0];            // -Fi
      v2f sB  = *(const v2f*)&STr[(kk * 20 + qq) * 20 + u0];
      v2f siB = *(const v2f*)&STi[(kk * 20 + qq) * 20 + u0];
      acc = wmma4(fA, sB, acc);    // + Fr*Sr
      acc = wmma4(fnA, siB, acc);  // - Fi*Si
    }
    // Output row stride is 20 floats per j; single guarded store per row.
    int pr0 = pt * 16 + 8 * (lane >> 4);
    size_t base = (size_t)bo * 8000u + (size_t)(kk * 400 + pr0 * 20 + qq);
#pragma unroll
    for (int j = 0; j < 8; ++j) {
      if (pr0 + j < 20) out[base + (size_t)(20 * j)] = acc[j] + bval;
    }
  }
}

// ---------------- host entry ----------------
extern "C" void kernel_launch(void* const* d_in, const int* in_sizes, int n_in,
                              void* d_out, int out_size, void* d_ws, size_t ws_size,
                              hipStream_t stream) {
  (void)in_sizes; (void)n_in; (void)out_size; (void)ws_size;
  const float* x    = (const float*)d_in[0];  // (512,2,60,60)
  const float* kern = (const float*)d_in[1];  // (2,5,6)
  const float* bias = (const float*)d_in[2];  // (5)
  float* out = (float*)d_out;                 // (512,5,20,20,20)
  float* ws  = (float*)d_ws;

  precompute_kernel<<<1, 256, 0, stream>>>(kern, ws);
  fhat_kernel<<<512, 256, 0, stream>>>(x, ws);
  s2conv_main_kernel<<<2560, 256, 0, stream>>>(ws, bias, out);
}